// SoftProjection_79688823210615
// MI455X (gfx1250) — compile-verified
//
#include <hip/hip_runtime.h>
#include <math.h>

typedef __attribute__((ext_vector_type(2))) float v2f;
typedef __attribute__((ext_vector_type(8))) float v8f;

#define BATCH 8
#define NPTS  16384
#define NSMP  1024
#define KNN   10
#define TS    16      // samples per block (one WMMA M-tile)
#define CHUNK 256     // points per chunk (8 waves x 2 x 16-pt N-tiles)
#define ROWPAD 260    // padded LDS row (260 % 64 == 4 -> rotating bank offset)

// async prefetch of one 16B slice per lane: global -> LDS, tracked by ASYNCcnt
__device__ __forceinline__ void prefetch_chunk(const float* xb, int base,
                                               float* dstbuf, int tid) {
  if (tid < (CHUNK * 3 * 4) / 16) {  // 192 lanes x b128 = 3072 B = 256 pts x 12 B
    unsigned lds = (unsigned)(size_t)dstbuf + (unsigned)tid * 16u;
    unsigned long long ga =
        (unsigned long long)(size_t)(xb + (size_t)base * 3) + (unsigned long long)tid * 16ull;
    asm volatile("global_load_async_to_lds_b128 %0, %1, off"
                 :: "v"(lds), "v"(ga) : "memory");
  }
}

__global__ __launch_bounds__(256) void softproj_kernel(
    const float* __restrict__ xyz,       // [B, N, 3]
    const float* __restrict__ sxyz,      // [B, S, 3]
    const float* __restrict__ temp_ptr,  // [1]
    float* __restrict__ out)             // [B*S*3 + 1]
{
  __shared__ float ssx[TS], ssy[TS], ssz[TS], q2s[TS];
  __shared__ float d2s[TS][ROWPAD];
  __shared__ float cand_d[TS][160];
  __shared__ int   cand_i[TS][160];
  __shared__ __align__(16) float pbuf[2][CHUNK * 3];  // double-buffered point tiles

  const int tid  = threadIdx.x;
  const int wave = tid >> 5;
  const int lane = tid & 31;
  const int hf   = lane >> 4;   // lane half: 0 -> K=0,1   1 -> K=2,3
  const int idx  = lane & 15;   // M (for A) / N (for B) index within tile

  const int b      = blockIdx.x >> 6;          // 64 sample-tiles per batch
  const int s_base = (blockIdx.x & 63) * TS;

  const float* xb = xyz  + (size_t)b * NPTS * 3;
  const float* sb = sxyz + (size_t)b * NSMP * 3;

  // ---- kick off async prefetch of chunk 0 while we set up ----
  prefetch_chunk(xb, 0, pbuf[0], tid);

  // ---- load 16-sample tile + query norms into LDS ----
  if (tid < TS) {
    float x = sb[(s_base + tid) * 3 + 0];
    float y = sb[(s_base + tid) * 3 + 1];
    float z = sb[(s_base + tid) * 3 + 2];
    ssx[tid] = x; ssy[tid] = y; ssz[tid] = z;
    q2s[tid] = x * x + y * y + z * z;
  }
  asm volatile("s_wait_asynccnt 0x0" ::: "memory");
  __syncthreads();

  // ---- A operand (16x4 f32): lanes0-15 hold (x,y)=K0,K1; lanes16-31 hold (z,0)=K2,K3 ----
  v2f a;
  a.x = hf ? ssz[idx] : ssx[idx];
  a.y = hf ? 0.0f     : ssy[idx];

  // per-lane copy of the 8 query norms this lane's C-components correspond to
  float q2r[8];
#pragma unroll
  for (int v = 0; v < 8; ++v) q2r[v] = q2s[v + 8 * hf];

  // ---- selector role: sample sm, strided slot; register-resident top-10 ----
  const int sm   = tid >> 4;
  const int slot = tid & 15;
  float bd[KNN]; int bi[KNN];
#pragma unroll
  for (int k = 0; k < KNN; ++k) { bd[k] = 3.0e38f; bi[k] = 0; }

  // ---- main loop: 64 chunks of 256 points, double-buffered via async copies ----
  int sel = 0;
  for (int base = 0; base < NPTS; base += CHUNK, sel ^= 1) {
    // prefetch next chunk into the other buffer (overlaps WMMA + selection)
    if (base + CHUNK < NPTS) prefetch_chunk(xb, base + CHUNK, pbuf[sel ^ 1], tid);

    const float* pb = pbuf[sel];

    // each wave computes two 16x16 distance tiles via WMMA
#pragma unroll
    for (int tt = 0; tt < 2; ++tt) {
      int n = (wave * 2 + tt) * 16 + idx;   // column within chunk
      float px = pb[n * 3 + 0];
      float py = pb[n * 3 + 1];
      float pz = pb[n * 3 + 2];
      float p2 = px * px + py * py + pz * pz;

      // B operand (4x16 f32), mirrored layout of A
      v2f bb;
      bb.x = hf ? pz   : px;
      bb.y = hf ? 0.0f : py;

      v8f c = {};
      // cross[m][n] = dot(sample_m, point_n)  -> v_wmma_f32_16x16x4_f32
      v8f d = __builtin_amdgcn_wmma_f32_16x16x4_f32(
          /*neg_a=*/false, a, /*neg_b=*/false, bb,
          /*c_mod=*/(short)0, c, /*reuse_a=*/false, /*reuse_b=*/false);

      // d2[m][n] = q2[m] + p2[n] - 2*cross, clamped at 0; transpose through LDS
#pragma unroll
      for (int v = 0; v < 8; ++v) {
        float dist = fmaxf(q2r[v] + p2 - 2.0f * d[v], 0.0f);
        d2s[v + 8 * hf][n] = dist;
      }
    }
    __syncthreads();

    // selection: 16 threads per sample, each scans 16 strided columns
#pragma unroll
    for (int cc = 0; cc < 16; ++cc) {
      int col  = slot + 16 * cc;
      float dv = d2s[sm][col];
      if (dv < bd[KNN - 1]) {
        bd[KNN - 1] = dv; bi[KNN - 1] = base + col;
#pragma unroll
        for (int k = KNN - 2; k >= 0; --k) {
          if (bd[k + 1] < bd[k]) {
            float td = bd[k]; bd[k] = bd[k + 1]; bd[k + 1] = td;
            int   ti = bi[k]; bi[k] = bi[k + 1]; bi[k + 1] = ti;
          }
        }
      }
    }
    // next chunk's buffer must be resident before anyone reads it next iteration
    asm volatile("s_wait_asynccnt 0x0" ::: "memory");
    __syncthreads();
  }

  // ---- dump 16 partial top-10 lists per sample into LDS ----
#pragma unroll
  for (int k = 0; k < KNN; ++k) {
    cand_d[sm][slot * KNN + k] = bd[k];
    cand_i[sm][slot * KNN + k] = bi[k];
  }
  __syncthreads();

  // ---- final merge + softmax + weighted centroid (one thread per sample) ----
  if (tid < TS) {
    const int m = tid;
    float fd[KNN]; int fi[KNN];
#pragma unroll
    for (int k = 0; k < KNN; ++k) {
      float best = 3.0e38f; int bestj = 0;
      for (int j = 0; j < 160; ++j) {
        float v = cand_d[m][j];
        if (v < best) { best = v; bestj = j; }
      }
      fd[k] = best;
      fi[k] = cand_i[m][bestj];
      cand_d[m][bestj] = 3.0e38f;   // mark consumed
    }

    float t    = temp_ptr[0];
    float t2   = t * t;
    float dmin = fd[0];             // extracted in ascending order
    float wsum = 0.0f, ox = 0.0f, oy = 0.0f, oz = 0.0f;
#pragma unroll
    for (int k = 0; k < KNN; ++k) {
      float w = expf(-(fd[k] - dmin) / t2);  // softmax with max-subtraction
      int   p = fi[k];
      ox += w * xb[p * 3 + 0];
      oy += w * xb[p * 3 + 1];
      oz += w * xb[p * 3 + 2];
      wsum += w;
    }
    float inv = 1.0f / wsum;
    size_t o = ((size_t)b * NSMP + (size_t)(s_base + m)) * 3;
    out[o + 0] = ox * inv;
    out[o + 1] = oy * inv;
    out[o + 2] = oz * inv;
  }

  // pass-through temp as second tuple output
  if (blockIdx.x == 0 && tid == 0) out[(size_t)BATCH * NSMP * 3] = temp_ptr[0];
}

extern "C" void kernel_launch(void* const* d_in, const int* in_sizes, int n_in,
                              void* d_out, int out_size, void* d_ws, size_t ws_size,
                              hipStream_t stream) {
  (void)in_sizes; (void)n_in; (void)d_ws; (void)ws_size; (void)out_size;
  const float* xyz      = (const float*)d_in[0];
  const float* sxyz     = (const float*)d_in[1];
  const float* temp_ptr = (const float*)d_in[2];
  float* out            = (float*)d_out;

  dim3 grid(BATCH * (NSMP / TS));   // 8 * 64 = 512 blocks
  dim3 block(256);                  // 8 wave32 per block
  softproj_kernel<<<grid, block, 0, stream>>>(xyz, sxyz, temp_ptr, out);
}